// ICP_12884901888484
// MI455X (gfx1250) — compile-verified
//
#include <hip/hip_runtime.h>
#include <math.h>

typedef __attribute__((ext_vector_type(2))) float v2f;
typedef __attribute__((ext_vector_type(8))) float v8f;

#define NPTS 4096
#define NB   4
#define STEPS 25          // STEPLIM + 1
#define ICP_TOL 0.0001f

// ---------------- workspace layout (byte offsets) ----------------
// scalars[0]=err carry, [1]=done flag, [2]=apply flag
#define WS_SCALARS   0                          // float[16]
#define WS_R         256                        // float[4][9]
#define WS_T         768                        // float[4][3]
#define WS_PART      1024                       // float[4][16]
#define WS_IDX       2048                       // int[4096]
#define WS_DMIN      (2048 + 4096*4)            // float[4*4096]
#define WS_TEMP      (WS_DMIN + 4*4096*4)       // float[4*4096*3]
#define WS_TPACK     (WS_TEMP + 4*4096*3*4)     // float4[4*4096]

// ================= init: temp = newpc, reset loop state =================
__global__ void icp_init(const float* __restrict__ newpc,
                         float* __restrict__ temp,
                         float* __restrict__ scalars) {
    int i = blockIdx.x * blockDim.x + threadIdx.x;
    if (i < NB * NPTS * 3) temp[i] = newpc[i];
    if (i == 0) { scalars[0] = 0.f; scalars[1] = 0.f; scalars[2] = 1.f; }
}

// ====== pack temp into float4 {x,y,z,|t|^2} for aligned b128 loads ======
__global__ void icp_pack(const float* __restrict__ temp,
                         float4* __restrict__ tpack) {
    int i = blockIdx.x * blockDim.x + threadIdx.x;
    if (i >= NB * NPTS) return;
    float x = temp[i*3], y = temp[i*3+1], z = temp[i*3+2];
    float4 p; p.x = x; p.y = y; p.z = z; p.w = x*x + y*y + z*z;
    tpack[i] = p;
}

// ========== distance matrix + row-min(+argmin) via f32 WMMA 16x16x4 ==========
// Per wave: one 16-row strip of one batch.
//   A' = [-2ox, -2oy, -2oz, 1]   (K=0,1 in lanes 0-15; K=2,3 in lanes 16-31)
//   B' = [tx; ty; tz; |t|^2]
//   C  = |o[m]|^2 per row (wave-constant, built once via shuffles)
//   D  = |o|^2 + |t|^2 - 2<o,t>  in one v_wmma_f32_16x16x4_f32 per tile.
// Clamp-to-0 is deferred to after the min reduction (monotone => identical).
__global__ __launch_bounds__(128)
void icp_dist_argmin(const float* __restrict__ origin,
                     const float4* __restrict__ tpack,
                     float* __restrict__ dmin,
                     int* __restrict__ idxout) {
    const int wave  = blockIdx.x * (blockDim.x >> 5) + (threadIdx.x >> 5);
    const int b     = wave >> 8;            // 256 strips per batch
    const int strip = wave & 255;
    const int mbase = strip << 4;
    const int lane  = threadIdx.x & 31;
    const int r     = lane & 15;
    const bool hi   = lane >= 16;

    const float* Op = origin + ((size_t)b * NPTS + mbase + r) * 3;
    float ox = Op[0], oy = Op[1], oz = Op[2];
    float osq = ox*ox + oy*oy + oz*oz;
    v2f a;
    a.x = hi ? (-2.f * oz) : (-2.f * ox);
    a.y = hi ? 1.f         : (-2.f * oy);

    // C[v] = |o|^2 of row (v + 8*hi): rows r live in lanes r and r+16
    v8f c0;
#pragma unroll
    for (int v = 0; v < 8; ++v)
        c0[v] = __shfl(osq, v + (hi ? 8 : 0), 32);

    float minv[8];
#pragma unroll
    for (int v = 0; v < 8; ++v) minv[v] = 3.4e38f;

    const float4* Tb = tpack + (size_t)b * NPTS;

    if (b == 3) {
        // ---- min + argmin path (reference takes idx from batch 3 only) ----
        int mini[8];
#pragma unroll
        for (int v = 0; v < 8; ++v) mini[v] = 0;
#pragma unroll 4
        for (int nc = 0; nc < NPTS / 16; ++nc) {
            const int n = (nc << 4) + r;
            float4 tp = Tb[n];
            v2f bv;
            bv.x = hi ? tp.z : tp.x;
            bv.y = hi ? tp.w : tp.y;
            v8f d = __builtin_amdgcn_wmma_f32_16x16x4_f32(
                false, a, false, bv, (short)0, c0, false, false);
#pragma unroll
            for (int v = 0; v < 8; ++v) {
                bool lt = d[v] < minv[v];
                minv[v] = lt ? d[v] : minv[v];
                mini[v] = lt ? n    : mini[v];
            }
        }
#pragma unroll
        for (int v = 0; v < 8; ++v) {
            float mv = minv[v]; int mi = mini[v];
#pragma unroll
            for (int off = 8; off >= 1; off >>= 1) {     // stays in 16-lane half
                float ov = __shfl_xor(mv, off, 32);
                int   oi = __shfl_xor(mi, off, 32);
                if (ov < mv || (ov == mv && oi < mi)) { mv = ov; mi = oi; }
            }
            minv[v] = mv; mini[v] = mi;
        }
        if (r == 0) {
#pragma unroll
            for (int v = 0; v < 8; ++v) {
                int m = mbase + v + (hi ? 8 : 0);        // C/D rows v / v+8
                dmin[b * NPTS + m] = sqrtf(fmaxf(minv[v], 0.f));
                idxout[m] = mini[v];
            }
        }
    } else {
        // ---- min-only path (batches 0..2) ----
#pragma unroll 4
        for (int nc = 0; nc < NPTS / 16; ++nc) {
            const int n = (nc << 4) + r;
            float4 tp = Tb[n];
            v2f bv;
            bv.x = hi ? tp.z : tp.x;
            bv.y = hi ? tp.w : tp.y;
            v8f d = __builtin_amdgcn_wmma_f32_16x16x4_f32(
                false, a, false, bv, (short)0, c0, false, false);
#pragma unroll
            for (int v = 0; v < 8; ++v)
                minv[v] = d[v] < minv[v] ? d[v] : minv[v];
        }
#pragma unroll
        for (int v = 0; v < 8; ++v) {
            float mv = minv[v];
#pragma unroll
            for (int off = 8; off >= 1; off >>= 1) {
                float ov = __shfl_xor(mv, off, 32);
                mv = ov < mv ? ov : mv;
            }
            minv[v] = mv;
        }
        if (r == 0) {
#pragma unroll
            for (int v = 0; v < 8; ++v) {
                int m = mbase + v + (hi ? 8 : 0);
                dmin[b * NPTS + m] = sqrtf(fmaxf(minv[v], 0.f));
            }
        }
    }
}

// ===== per-batch reductions: sum p1, sum p2, sum p2_d*p1_e, sum dmin =====
// iteration: p1 = temp, p2[i] = origin[idx[i]] (gather, idx from batch 3)
// final:     p1 = newpc, p2 = temp (idx = null, dmin = null)
__global__ __launch_bounds__(256)
void icp_sums(const float* __restrict__ p1, const float* __restrict__ p2,
              const int* __restrict__ idx, const float* __restrict__ dmin,
              float* __restrict__ part) {
    __shared__ float red[256];
    const int b = blockIdx.x, tid = threadIdx.x;
    float acc[16];
#pragma unroll
    for (int k = 0; k < 16; ++k) acc[k] = 0.f;
    const float* P1 = p1 + (size_t)b * NPTS * 3;
    const float* P2 = p2 + (size_t)b * NPTS * 3;
    for (int i = tid; i < NPTS; i += 256) {
        float ax = P1[i*3], ay = P1[i*3+1], az = P1[i*3+2];
        int j = idx ? idx[i] : i;
        float bx = P2[j*3], by = P2[j*3+1], bz = P2[j*3+2];
        acc[0] += ax;  acc[1] += ay;  acc[2] += az;
        acc[3] += bx;  acc[4] += by;  acc[5] += bz;
        acc[6]  += bx*ax; acc[7]  += bx*ay; acc[8]  += bx*az;
        acc[9]  += by*ax; acc[10] += by*ay; acc[11] += by*az;
        acc[12] += bz*ax; acc[13] += bz*ay; acc[14] += bz*az;
        if (dmin) acc[15] += dmin[b * NPTS + i];
    }
    for (int k = 0; k < 16; ++k) {
        red[tid] = acc[k];
        __syncthreads();
        for (int s = 128; s > 0; s >>= 1) {
            if (tid < s) red[tid] += red[tid + s];
            __syncthreads();
        }
        if (tid == 0) part[b * 16 + k] = red[0];
        __syncthreads();
    }
}

// =================== 3x3 Kabsch helpers (scalar, per batch) ===================
__device__ inline float det3(const float M[3][3]) {
    return M[0][0]*(M[1][1]*M[2][2] - M[1][2]*M[2][1])
         - M[0][1]*(M[1][0]*M[2][2] - M[1][2]*M[2][0])
         + M[0][2]*(M[1][0]*M[2][1] - M[1][1]*M[2][0]);
}

__device__ inline void jrot(float B[3][3], float V[3][3], int p, int q) {
    float apq = B[p][q];
    if (fabsf(apq) < 1e-30f) return;
    float tau = (B[q][q] - B[p][p]) / (2.f * apq);
    float tt  = (tau >= 0.f ? 1.f : -1.f) / (fabsf(tau) + sqrtf(1.f + tau*tau));
    float c   = 1.f / sqrtf(1.f + tt*tt);
    float s   = tt * c;
    for (int k = 0; k < 3; ++k) {               // B = B * J
        float bkp = B[k][p], bkq = B[k][q];
        B[k][p] = c*bkp - s*bkq;
        B[k][q] = s*bkp + c*bkq;
    }
    for (int k = 0; k < 3; ++k) {               // B = J^T * B
        float bpk = B[p][k], bqk = B[q][k];
        B[p][k] = c*bpk - s*bqk;
        B[q][k] = s*bpk + c*bqk;
    }
    for (int k = 0; k < 3; ++k) {               // V = V * J
        float vkp = V[k][p], vkq = V[k][q];
        V[k][p] = c*vkp - s*vkq;
        V[k][q] = s*vkp + c*vkq;
    }
}

// R for get_transform: H = U S V^T ; d = det(V)det(U) ; V'[2][2] = V[2][2]*d ;
// R = V' U^T   (reference scales only element (2,2) of v)
__device__ void kabschR(const float H[3][3], float R[3][3]) {
    float B[3][3], V[3][3] = {{1,0,0},{0,1,0},{0,0,1}};
    for (int i = 0; i < 3; ++i)
        for (int j = 0; j < 3; ++j) {
            float s = 0.f;
            for (int k = 0; k < 3; ++k) s += H[k][i] * H[k][j];  // H^T H
            B[i][j] = s;
        }
    for (int sw = 0; sw < 16; ++sw) { jrot(B,V,0,1); jrot(B,V,0,2); jrot(B,V,1,2); }
    float lam[3] = { B[0][0], B[1][1], B[2][2] };
    for (int a = 0; a < 2; ++a)
        for (int c2i = a + 1; c2i < 3; ++c2i)
            if (lam[c2i] > lam[a]) {
                float tl = lam[a]; lam[a] = lam[c2i]; lam[c2i] = tl;
                for (int k = 0; k < 3; ++k) {
                    float tv = V[k][a]; V[k][a] = V[k][c2i]; V[k][c2i] = tv;
                }
            }
    float u0[3], u1[3], u2[3];
    for (int k = 0; k < 3; ++k)
        u0[k] = H[k][0]*V[0][0] + H[k][1]*V[1][0] + H[k][2]*V[2][0];
    float n0 = sqrtf(u0[0]*u0[0] + u0[1]*u0[1] + u0[2]*u0[2]);
    if (n0 > 1e-18f) { float rr = 1.f/n0; u0[0]*=rr; u0[1]*=rr; u0[2]*=rr; }
    else { u0[0] = 1.f; u0[1] = 0.f; u0[2] = 0.f; }
    for (int k = 0; k < 3; ++k)
        u1[k] = H[k][0]*V[0][1] + H[k][1]*V[1][1] + H[k][2]*V[2][1];
    float dp = u1[0]*u0[0] + u1[1]*u0[1] + u1[2]*u0[2];
    for (int k = 0; k < 3; ++k) u1[k] -= dp * u0[k];
    float n1 = sqrtf(u1[0]*u1[0] + u1[1]*u1[1] + u1[2]*u1[2]);
    if (n1 > 1e-18f) { float rr = 1.f/n1; u1[0]*=rr; u1[1]*=rr; u1[2]*=rr; }
    else {
        float ax = (fabsf(u0[0]) < 0.9f) ? 1.f : 0.f;
        float ay = 1.f - ax;
        u1[0] = u0[1]*0.f - u0[2]*ay;
        u1[1] = u0[2]*ax  - u0[0]*0.f;
        u1[2] = u0[0]*ay  - u0[1]*ax;
        float nn = sqrtf(u1[0]*u1[0] + u1[1]*u1[1] + u1[2]*u1[2]) + 1e-30f;
        u1[0] /= nn; u1[1] /= nn; u1[2] /= nn;
    }
    u2[0] = u0[1]*u1[2] - u0[2]*u1[1];
    u2[1] = u0[2]*u1[0] - u0[0]*u1[2];
    u2[2] = u0[0]*u1[1] - u0[1]*u1[0];
    float sH = (det3(H) >= 0.f) ? 1.f : -1.f;
    float sV = (det3(V) >= 0.f) ? 1.f : -1.f;
    float sU = sH * sV;                          // det(U)
    u2[0] *= sU; u2[1] *= sU; u2[2] *= sU;
    float d = sV * sU;                           // det(V)*det(U)
    float Vp[3][3];
    for (int i = 0; i < 3; ++i) for (int j = 0; j < 3; ++j) Vp[i][j] = V[i][j];
    Vp[2][2] = V[2][2] * d;
    float U[3][3] = { { u0[0], u1[0], u2[0] },
                      { u0[1], u1[1], u2[1] },
                      { u0[2], u1[2], u2[2] } };
    for (int i = 0; i < 3; ++i)
        for (int j = 0; j < 3; ++j)              // R = Vp * U^T
            R[i][j] = Vp[i][0]*U[j][0] + Vp[i][1]*U[j][1] + Vp[i][2]*U[j][2];
}

__device__ void partToRt(const float* P, float R[3][3], float t[3]) {
    float c1[3], c2[3], H[3][3];
    for (int k = 0; k < 3; ++k) {
        c1[k] = P[k]     * (1.f / NPTS);
        c2[k] = P[3 + k] * (1.f / NPTS);
    }
    for (int d0 = 0; d0 < 3; ++d0)
        for (int e = 0; e < 3; ++e)
            H[d0][e] = P[6 + d0*3 + e] - (float)NPTS * c2[d0] * c1[e];
    kabschR(H, R);
    for (int e = 0; e < 3; ++e)
        t[e] = c2[e] - (c1[0]*R[0][e] + c1[1]*R[1][e] + c1[2]*R[2][e]);
}

// ========== per-iteration: err/done update + per-batch R,t ==========
__global__ void icp_transform(float* __restrict__ scalars,
                              const float* __restrict__ part,
                              float* __restrict__ Rbuf,
                              float* __restrict__ tbuf) {
    int tid = threadIdx.x;
    if (tid == 0) {
        float errsum = part[15] + part[31] + part[47] + part[63];
        float errnew = errsum * (1.f / (float)(NB * NPTS));
        float err  = scalars[0];
        bool  done = scalars[1] > 0.5f;
        bool  nd   = done || (fabsf(err - errnew) < ICP_TOL);
        scalars[0] = done ? err : errnew;
        scalars[1] = nd   ? 1.f : 0.f;
        scalars[2] = done ? 0.f : 1.f;           // apply uses OLD done flag
    }
    if (tid < NB) {
        float R[3][3], t[3];
        partToRt(part + tid * 16, R, t);
        for (int i = 0; i < 3; ++i)
            for (int j = 0; j < 3; ++j) Rbuf[tid*9 + i*3 + j] = R[i][j];
        for (int e = 0; e < 3; ++e) tbuf[tid*3 + e] = t[e];
    }
}

// ========== temp = temp @ R + t (skipped once converged) ==========
__global__ void icp_apply(float* __restrict__ temp,
                          const float* __restrict__ Rbuf,
                          const float* __restrict__ tbuf,
                          const float* __restrict__ scalars) {
    int i = blockIdx.x * blockDim.x + threadIdx.x;
    if (i >= NB * NPTS) return;
    if (scalars[2] < 0.5f) return;
    int b = i >> 12;
    const float* Rb = Rbuf + b * 9;
    const float* tb = tbuf + b * 3;
    float* p = temp + (size_t)i * 3;
    float x = p[0], y = p[1], z = p[2];
    float nx = x*Rb[0] + y*Rb[3] + z*Rb[6] + tb[0];
    float ny = x*Rb[1] + y*Rb[4] + z*Rb[7] + tb[1];
    float nz = x*Rb[2] + y*Rb[5] + z*Rb[8] + tb[2];
    p[0] = nx; p[1] = ny; p[2] = nz;
}

// ========== final: R,t for (newpc -> temp), quaternion, pack output ==========
__device__ void rotToQuat(const float m[3][3], float q[4]) {
    float m00=m[0][0], m01=m[0][1], m02=m[0][2];
    float m10=m[1][0], m11=m[1][1], m12=m[1][2];
    float m20=m[2][0], m21=m[2][1], m22=m[2][2];
    float tr = m00 + m11 + m22;
    if (tr > 0.f) {
        float s0 = sqrtf(fmaxf(tr + 1.f, 1e-12f)) * 2.f;
        q[0] = (m21 - m12) / s0; q[1] = (m02 - m20) / s0;
        q[2] = (m10 - m01) / s0; q[3] = s0 * 0.25f;
    } else if (m00 >= m11 && m00 >= m22) {
        float s1 = sqrtf(fmaxf(1.f + m00 - m11 - m22, 1e-12f)) * 2.f;
        q[0] = s1 * 0.25f;       q[1] = (m01 + m10) / s1;
        q[2] = (m02 + m20) / s1; q[3] = (m21 - m12) / s1;
    } else if (m11 >= m22) {
        float s2 = sqrtf(fmaxf(1.f - m00 + m11 - m22, 1e-12f)) * 2.f;
        q[0] = (m01 + m10) / s2; q[1] = s2 * 0.25f;
        q[2] = (m12 + m21) / s2; q[3] = (m02 - m20) / s2;
    } else {
        float s3 = sqrtf(fmaxf(1.f - m00 - m11 + m22, 1e-12f)) * 2.f;
        q[0] = (m02 + m20) / s3; q[1] = (m12 + m21) / s3;
        q[2] = s3 * 0.25f;       q[3] = (m10 - m01) / s3;
    }
}

__global__ void icp_finalize(const float* __restrict__ part,
                             float* __restrict__ out) {
    int tid = threadIdx.x;
    if (tid >= NB) return;
    float R[3][3], t[3], q[4];
    partToRt(part + tid * 16, R, t);
    rotToQuat(R, q);
    out[tid*7 + 0] = t[0]; out[tid*7 + 1] = t[1]; out[tid*7 + 2] = t[2];
    out[tid*7 + 3] = q[0]; out[tid*7 + 4] = q[1];
    out[tid*7 + 5] = q[2]; out[tid*7 + 6] = q[3];
}

// =============================== launch ===============================
extern "C" void kernel_launch(void* const* d_in, const int* in_sizes, int n_in,
                              void* d_out, int out_size, void* d_ws, size_t ws_size,
                              hipStream_t stream) {
    (void)in_sizes; (void)n_in; (void)out_size; (void)ws_size;
    const float* newpc    = (const float*)d_in[0];
    const float* originpc = (const float*)d_in[1];
    char* ws = (char*)d_ws;
    float*  scalars = (float*)(ws + WS_SCALARS);
    float*  Rbuf    = (float*)(ws + WS_R);
    float*  tbuf    = (float*)(ws + WS_T);
    float*  part    = (float*)(ws + WS_PART);
    int*    idx     = (int*)  (ws + WS_IDX);
    float*  dmin    = (float*)(ws + WS_DMIN);
    float*  temp    = (float*)(ws + WS_TEMP);
    float4* tpack   = (float4*)(ws + WS_TPACK);
    float*  out     = (float*)d_out;

    icp_init<<<(NB*NPTS*3 + 255) / 256, 256, 0, stream>>>(newpc, temp, scalars);
    icp_pack<<<(NB*NPTS + 255) / 256, 256, 0, stream>>>(temp, tpack);

    for (int it = 0; it < STEPS; ++it) {
        icp_dist_argmin<<<256, 128, 0, stream>>>(originpc, tpack, dmin, idx);
        icp_sums<<<NB, 256, 0, stream>>>(temp, originpc, idx, dmin, part);
        icp_transform<<<1, 32, 0, stream>>>(scalars, part, Rbuf, tbuf);
        icp_apply<<<(NB*NPTS + 255) / 256, 256, 0, stream>>>(temp, Rbuf, tbuf, scalars);
        icp_pack<<<(NB*NPTS + 255) / 256, 256, 0, stream>>>(temp, tpack);
    }

    icp_sums<<<NB, 256, 0, stream>>>(newpc, temp, (const int*)nullptr,
                                     (const float*)nullptr, part);
    icp_finalize<<<1, 32, 0, stream>>>(part, out);
}